// OnnxNonMaxSuppression_12678743458388
// MI455X (gfx1250) — compile-verified
//
#include <hip/hip_runtime.h>
#include <stdint.h>

// NonMaxSuppression: B=8 batches, C=16 classes, N=2048 boxes, MAX_OUT=100.
// One workgroup per (b,c). Boxes for batch b staged to LDS with CDNA5
// async global->LDS DMA. Greedy NMS as iterative (argmax -> suppress),
// which is exactly equivalent to the reference's scan over stable-sorted order.

static constexpr int NB  = 8;
static constexpr int NC  = 16;
static constexpr int NN  = 2048;
static constexpr int TPB = 256;
static constexpr int EPT = NN / TPB;   // 8 elements per thread

// Monotone map: float -> uint32 preserving total order (handles negatives too).
__device__ __forceinline__ uint32_t f32_ord(float f) {
    uint32_t b = __float_as_uint(f);
    return b ^ (uint32_t)((int32_t(b) >> 31) | 0x80000000);
}

__global__ __launch_bounds__(TPB) void nms_kernel(
    const float* __restrict__ boxes,     // [NB, NN, 4] (x1,y1,x2,y2)
    const float* __restrict__ scores,    // [NB, NC, NN]
    const float* __restrict__ iou_t_p,   // [1]
    const float* __restrict__ score_t_p, // [1]
    int*         __restrict__ out,       // [NB*NC*MO, 3] int32
    int MO)
{
    __shared__ __align__(16) float sBoxes[NN * 4];      // 32 KB
    __shared__ unsigned long long sWave[TPB / 32];
    __shared__ unsigned long long sWin;

    const int c = blockIdx.x;
    const int b = blockIdx.y;
    const int t = threadIdx.x;

    // ---------- Stage boxes[b,:,:] into LDS via async global->LDS DMA ----------
    {
        uint32_t lds_base = (uint32_t)(uintptr_t)&sBoxes[0];   // low 32 bits of
                                                               // LDS flat addr == LDS offset
        const char* gbase = (const char*)(boxes + (size_t)b * NN * 4);
        #pragma unroll
        for (int j = 0; j < EPT; ++j) {
            uint32_t off  = (uint32_t)(t + j * TPB) * 16u;     // 16B per lane per issue
            uint32_t ldsa = lds_base + off;
            uint64_t ga   = (uint64_t)(uintptr_t)(gbase + off);
            asm volatile("global_load_async_to_lds_b128 %0, %1, off"
                         :
                         : "v"(ldsa), "v"(ga)
                         : "memory");
        }
        asm volatile("s_wait_asynccnt 0" ::: "memory");
    }
    __syncthreads();

    const float iou_t   = *iou_t_p;
    const float score_t = *score_t_p;

    // ---------- Per-thread register state: 8 elements, strided by TPB ----------
    float bx1[EPT], by1[EPT], bx2[EPT], by2[EPT], ar[EPT];
    unsigned long long key[EPT];   // 0 == inactive (suppressed / below threshold)

    const float* sc = scores + ((size_t)b * NC + c) * NN;
    #pragma unroll
    for (int j = 0; j < EPT; ++j) {
        int idx = j * TPB + t;
        float x1 = sBoxes[idx * 4 + 0];
        float y1 = sBoxes[idx * 4 + 1];
        float x2 = sBoxes[idx * 4 + 2];
        float y2 = sBoxes[idx * 4 + 3];
        bx1[j] = x1; by1[j] = y1; bx2[j] = x2; by2[j] = y2;
        ar[j]  = (x2 - x1) * (y2 - y1);
        float s = sc[idx];                                   // coalesced
        bool active = (s > score_t);
        // hi: score (order-preserving), lo: NN-1-idx so ties pick smallest index,
        // matching stable argsort(-scores).
        key[j] = active
               ? (((unsigned long long)f32_ord(s) << 32) |
                  (unsigned long long)(uint32_t)(NN - 1 - idx))
               : 0ull;
    }

    // ---------- Greedy NMS: iterative argmax + suppress ----------
    const size_t row_base = ((size_t)(b * NC + c)) * (size_t)MO;
    int kept = 0;
    for (int k = 0; k < MO; ++k) {
        // local max over 8 register elements
        unsigned long long m = 0ull;
        #pragma unroll
        for (int j = 0; j < EPT; ++j) m = (key[j] > m) ? key[j] : m;
        // wave32 reduction
        #pragma unroll
        for (int off = 16; off >= 1; off >>= 1) {
            unsigned long long o = __shfl_xor(m, off, 32);
            m = (o > m) ? o : m;
        }
        if ((t & 31) == 0) sWave[t >> 5] = m;
        __syncthreads();
        if (t == 0) {
            unsigned long long w = sWave[0];
            #pragma unroll
            for (int i = 1; i < TPB / 32; ++i)
                w = (sWave[i] > w) ? sWave[i] : w;
            sWin = w;
        }
        __syncthreads();
        unsigned long long w = sWin;          // uniform across block
        if (w == 0ull) break;                 // no remaining candidate

        int widx = (NN - 1) - (int)(uint32_t)(w & 0xFFFFFFFFu);
        if (t == 0) {
            int* row = out + (row_base + (size_t)k) * 3;
            row[0] = b; row[1] = c; row[2] = widx;
        }
        kept = k + 1;

        // winner box broadcast-read from LDS (all lanes same address)
        float wx1 = sBoxes[widx * 4 + 0];
        float wy1 = sBoxes[widx * 4 + 1];
        float wx2 = sBoxes[widx * 4 + 2];
        float wy2 = sBoxes[widx * 4 + 3];
        float war = (wx2 - wx1) * (wy2 - wy1);

        #pragma unroll
        for (int j = 0; j < EPT; ++j) {
            if (key[j] == 0ull) continue;
            int idx = j * TPB + t;
            if (idx == widx) { key[j] = 0ull; continue; }
            float iw = fminf(bx2[j], wx2) - fmaxf(bx1[j], wx1);
            float ih = fminf(by2[j], wy2) - fmaxf(by1[j], wy1);
            iw = fmaxf(iw, 0.0f);
            ih = fmaxf(ih, 0.0f);
            float inter = iw * ih;
            float iou   = inter / (ar[j] + war - inter);
            if (iou > iou_t) key[j] = 0ull;
        }
    }

    // ---------- Fill unselected rows with -1 (parallel) ----------
    for (int r = kept + t; r < MO; r += TPB) {
        int* row = out + (row_base + (size_t)r) * 3;
        row[0] = -1; row[1] = -1; row[2] = -1;
    }
}

extern "C" void kernel_launch(void* const* d_in, const int* in_sizes, int n_in,
                              void* d_out, int out_size, void* d_ws, size_t ws_size,
                              hipStream_t stream) {
    const float* boxes   = (const float*)d_in[0];
    const float* scores  = (const float*)d_in[1];
    // d_in[2] = max_output_boxes_per_class (device); MO is recovered from out_size
    const float* iou_t   = (const float*)d_in[3];
    const float* score_t = (const float*)d_in[4];

    int MO = out_size / (NB * NC * 3);     // = 100

    dim3 grid(NC, NB);
    nms_kernel<<<grid, TPB, 0, stream>>>(boxes, scores, iou_t, score_t,
                                         (int*)d_out, MO);
}